// NormConvTranspose2d_14259291423054
// MI455X (gfx1250) — compile-verified
//
#include <hip/hip_runtime.h>

typedef __attribute__((ext_vector_type(2))) float v2f;
typedef __attribute__((ext_vector_type(8))) float v8f;

#define B_   2
#define C_   32
#define H_   96
#define W_   96
#define O_   48
#define HO_  192
#define WO_  192
#define KP_  (C_ * 2)          // 64 K-pairs (K = 128)
#define WN_CLS (KP_ * O_)      // 3072 float2 per (hc,wc) class (24 KB)

// tap tables: kh candidates per h-class, and validity per class
// hc: 0 = ho==0, 1 = odd interior, 2 = even interior, 3 = ho==191
__device__ __constant__ int d_ktab[4][2] = {{1,3},{0,2},{1,3},{0,2}};
__device__ __constant__ int d_kval[4][2] = {{1,0},{1,1},{1,1},{0,1}};

// K ordering convention (matched between A and B):
//   K = c*4 + th*2 + s, where s=0 is the tap at x[..][iw]   (kw = ktab[wc][1])
//                         s=1 is the tap at x[..][iw+1] (kw = ktab[wc][0])
// so an unswapped b64 load of x[iw], x[iw+1] is the A fragment directly.

// ---------------------------------------------------------------------------
// Kernel 1: normalized weights as interleaved K-pairs:
//   wn2[cls][c*2+th][o] = float2{ w(th, kw=ktab[wc][1]), w(th, kw=ktab[wc][0]) } / norm
// ---------------------------------------------------------------------------
__global__ void prep_weights(const float* __restrict__ w, v2f* __restrict__ wn2) {
    int tid = blockIdx.x * blockDim.x + threadIdx.x;
    if (tid >= O_ * C_ * 16) return;
    int cls = tid & 15;
    int c   = (tid >> 4) % C_;
    int o   = tid / (16 * C_);
    int hc = cls >> 2, wc = cls & 3;

    float wv[2][2];
    float norm = 0.f;
    for (int th = 0; th < 2; ++th)
        for (int tw = 0; tw < 2; ++tw) {
            float val = w[((o * C_ + c) * 4 + d_ktab[hc][th]) * 4 + d_ktab[wc][tw]];
            bool v = d_kval[hc][th] && d_kval[wc][tw];
            wv[th][tw] = v ? val : 0.f;
            norm      += v ? val : 0.f;
        }
    float inv = 1.f / norm;   // weights ~ U(0,1): norm > 0
    for (int th = 0; th < 2; ++th) {
        v2f p;
        p.x = wv[th][1] * inv;   // s=0: larger kw tap (x at iw)
        p.y = wv[th][0] * inv;   // s=1: smaller kw tap (x at iw+1)
        wn2[((size_t)cls * KP_ + c * 2 + th) * O_ + o] = p;
    }
}

// ---------------------------------------------------------------------------
// Kernel 2: interior pixels (ho,wo in [1,190]) as implicit GEMM with
//   V_WMMA_F32_16X16X4_F32.  One wave = 16 output pixels x 48 outputs (3 N-tiles),
//   K accumulated 4 at a time (one channel's 2x2 taps per WMMA).
//   A lanes 0-15: th=0 (K pair c*2+0); lanes 16-31: th=1 (K pair c*2+1).
// ---------------------------------------------------------------------------
__global__ void interior_gemm(const float* __restrict__ x, const v2f* __restrict__ wn2,
                              const float* __restrict__ bias, float* __restrict__ out) {
    __shared__ v2f ldsB[WN_CLS];   // 24 KB: interleaved B pairs for this class

    const int ph = blockIdx.y >> 1;   // 0: even ho, 1: odd ho
    const int pw = blockIdx.y & 1;    // 0: even wo, 1: odd wo
    const int hc = ph ? 1 : 2;
    const int wc = pw ? 1 : 2;

    const v2f* wnc = wn2 + (size_t)(hc * 4 + wc) * WN_CLS;
    for (int i = threadIdx.x; i < WN_CLS; i += blockDim.x) ldsB[i] = wnc[i];
    __syncthreads();

    const int wave = threadIdx.x >> 5;     // wave32
    const int lane = threadIdx.x & 31;
    const int t = blockIdx.x * 8 + wave;   // M-tile index
    if (t >= B_ * 95 * 6) return;          // uniform per-wave exit (EXEC stays all-1)

    const int b     = t / (95 * 6);
    const int rem   = t % (95 * 6);
    const int hoIdx = rem / 6;             // 0..94
    const int wTile = rem % 6;             // 16 pixels each over 95 columns
    const int ho    = (ph ? 1 : 2) + 2 * hoIdx;
    const int wofs  = pw ? 1 : 2;

    const int m  = lane & 15;              // M row within tile
    const int hi = lane >> 4;              // 0 -> th=0 (K0,K1), 1 -> th=1 (K2,K3)
    int wIdxRaw = wTile * 16 + m;
    int wIdx    = wIdxRaw > 94 ? 94 : wIdxRaw;   // clamp; extra rows never stored
    const int ihL = 1 + hoIdx - hi;        // ih for this lane's th (interior: in range)
    // x row pointer at (b, c=0, ihL, iw=wIdx): [0] = s=0 tap, [1] = s=1 tap
    const float* xrow = x + ((size_t)(b * C_) * H_ + ihL) * W_ + wIdx;

    v8f acc0 = {}, acc1 = {}, acc2 = {};
#pragma unroll
    for (int c = 0; c < C_; ++c) {
        v2f a;
        a.x = xrow[0];                     // K even (s=0)
        a.y = xrow[1];                     // K odd  (s=1)
        xrow += H_ * W_;
        const int kb = (c * 2 + hi) * O_ + m;   // interleaved pair index
        v2f b0 = ldsB[kb +  0];
        v2f b1 = ldsB[kb + 16];
        v2f b2 = ldsB[kb + 32];
        acc0 = __builtin_amdgcn_wmma_f32_16x16x4_f32(false, a, false, b0, (short)0, acc0, false, false);
        acc1 = __builtin_amdgcn_wmma_f32_16x16x4_f32(false, a, false, b1, (short)0, acc1, false, false);
        acc2 = __builtin_amdgcn_wmma_f32_16x16x4_f32(false, a, false, b2, (short)0, acc2, false, false);
    }

    // C/D layout: lanes 0-15 VGPR r -> (M=r, N=lane); lanes 16-31 -> (M=r+8, N=lane-16)
    const float bia0 = bias[m], bia1 = bias[16 + m], bia2 = bias[32 + m];
#pragma unroll
    for (int r = 0; r < 8; ++r) {
        int mRow  = hi * 8 + r;
        int wIdxS = wTile * 16 + mRow;
        if (wIdxS <= 94) {
            int wo = wofs + 2 * wIdxS;
            size_t base = (((size_t)b * O_ + m) * HO_ + ho) * WO_ + wo;
            out[base                         ] = acc0[r] + bia0;
            out[base + (size_t)16 * HO_ * WO_] = acc1[r] + bia1;
            out[base + (size_t)32 * HO_ * WO_] = acc2[r] + bia2;
        }
    }
}

// ---------------------------------------------------------------------------
// Kernel 3: border pixels (ho or wo in {0,191}); 764 pixels per (b,o) plane.
// ---------------------------------------------------------------------------
__global__ void border_kernel(const float* __restrict__ x, const v2f* __restrict__ wn2,
                              const float* __restrict__ bias, float* __restrict__ out) {
    int tid = blockIdx.x * blockDim.x + threadIdx.x;
    const int NB = 192 + 192 + 190 + 190;   // 764
    if (tid >= B_ * O_ * NB) return;
    int p  = tid % NB;
    int bo = tid / NB;
    int o  = bo % O_;
    int b  = bo / O_;

    int ho, wo;
    if      (p < 192) { ho = 0;   wo = p; }
    else if (p < 384) { ho = 191; wo = p - 192; }
    else if (p < 574) { wo = 0;   ho = 1 + (p - 384); }
    else              { wo = 191; ho = 1 + (p - 574); }

    int hc = (ho == 0) ? 0 : (ho == 191) ? 3 : ((ho & 1) ? 1 : 2);
    int wc = (wo == 0) ? 0 : (wo == 191) ? 3 : ((wo & 1) ? 1 : 2);

    int ih[2];
    for (int th = 0; th < 2; ++th) {
        int v = (ho + 1 - d_ktab[hc][th]) / 2;
        ih[th] = v < 0 ? 0 : (v > 95 ? 95 : v);    // invalid taps have wn == 0
    }
    // s=0 tap: kw = ktab[wc][1]; s=1 tap: kw = ktab[wc][0]  (matches prep layout)
    int v0 = (wo + 1 - d_ktab[wc][1]) / 2;
    int v1 = (wo + 1 - d_ktab[wc][0]) / 2;
    int iw0 = v0 < 0 ? 0 : (v0 > 95 ? 95 : v0);
    int iw1 = v1 < 0 ? 0 : (v1 > 95 ? 95 : v1);

    const v2f* wnc = wn2 + (size_t)(hc * 4 + wc) * WN_CLS;
    float sum = 0.f;
    for (int c = 0; c < C_; ++c) {
        const float* xc = x + ((size_t)(b * C_ + c) * H_) * W_;
        for (int th = 0; th < 2; ++th) {
            v2f pr = wnc[(c * 2 + th) * O_ + o];
            sum += xc[ih[th] * W_ + iw0] * pr.x;
            sum += xc[ih[th] * W_ + iw1] * pr.y;
        }
    }
    out[(((size_t)b * O_ + o) * HO_ + ho) * WO_ + wo] = sum + bias[o];
}

// ---------------------------------------------------------------------------
extern "C" void kernel_launch(void* const* d_in, const int* in_sizes, int n_in,
                              void* d_out, int out_size, void* d_ws, size_t ws_size,
                              hipStream_t stream) {
    const float* x    = (const float*)d_in[0];
    const float* w    = (const float*)d_in[1];
    const float* bias = (const float*)d_in[2];
    float* out = (float*)d_out;
    v2f* wn2   = (v2f*)d_ws;   // 16 * 64 * 48 float2 = 384 KB

    // 1) normalized weights per (o, c, 16 border/parity classes), interleaved pairs
    int np = O_ * C_ * 16;
    prep_weights<<<(np + 255) / 256, 256, 0, stream>>>(w, wn2);

    // 2) interior implicit GEMM (f32 WMMA): grid.y = 4 parity classes,
    //    8 waves/block, 1140 M-tiles per class -> 143 blocks
    int tiles  = B_ * 95 * 6;
    dim3 grid((tiles + 7) / 8, 4, 1);
    interior_gemm<<<grid, 256, 0, stream>>>(x, wn2, bias, out);

    // 3) border pixels
    int nb = B_ * O_ * (192 + 192 + 190 + 190);
    border_kernel<<<(nb + 255) / 256, 256, 0, stream>>>(x, wn2, bias, out);
}